// LSTM_69166153334834
// MI455X (gfx1250) — compile-verified
//
#include <hip/hip_runtime.h>
#include <hip/hip_bf16.h>

typedef __attribute__((ext_vector_type(2))) float v2f;
typedef __attribute__((ext_vector_type(8))) float v8f;

#define T_LEN 512
#define BATCH 64
#define HID   1024
#define N4    4096   // 4*HID
#define K2    2048   // 2*HID
#define KC    128    // K chunk staged in LDS
#define AST   132    // A LDS stride (KC + 4 pad -> bank step 4, conflict-free frags)
#define BST   132    // B^T LDS stride (KC + 4 pad)
#define GST   20     // gate LDS row stride (16 + 4 pad)

// ---------------------------------------------------------------------------
// Phase 1 (optional, needs 512MB ws): XG[t*B+m, n] = x2d @ Wx0 + b0
//   x2d [32768, 1024] @ W [1024, 4096]; block computes a 64x64 tile.
//   4 waves, each with a 2x2 register tile of 16x16 WMMA accumulators.
// ---------------------------------------------------------------------------
__global__ __launch_bounds__(128) void xw_gemm_kernel(
    const float* __restrict__ x2d,   // [32768, 1024]
    const float* __restrict__ W,     // [1024, 4096] (layer-0 x-weights)
    const float* __restrict__ bias,  // [4096]
    float*       __restrict__ XG)    // [32768, 4096]
{
  __shared__ float Alds[64 * AST];
  __shared__ float Blds[64 * BST];   // transposed: [n-col][k]

  const int tid  = threadIdx.x;
  const int lane = tid & 31;
  const int wave = tid >> 5;         // 0..3
  const int half = lane >> 4;
  const int lp   = lane & 15;
  const int mt0  = (wave & 1) * 2;   // M tiles {mt0, mt0+1}
  const int nt0  = (wave >> 1) * 2;  // N tiles {nt0, nt0+1}
  const int n0   = blockIdx.x * 64;
  const size_t r0 = (size_t)blockIdx.y * 64;

  v8f acc[2][2] = {};

  for (int kb = 0; kb < HID; kb += KC) {
    for (int i = tid; i < 64 * (KC / 4); i += 128) {
      int row = i >> 5, c4 = i & 31;
      float4 v = *(const float4*)(x2d + (r0 + row) * HID + kb + c4 * 4);
      *(float4*)(&Alds[row * AST + c4 * 4]) = v;
    }
    for (int i = tid; i < KC * 16; i += 128) {
      int kk = i >> 4, c4 = i & 15;
      float4 v = *(const float4*)(W + (size_t)(kb + kk) * N4 + n0 + c4 * 4);
      int cb = c4 * 4;
      Blds[(cb + 0) * BST + kk] = v.x;
      Blds[(cb + 1) * BST + kk] = v.y;
      Blds[(cb + 2) * BST + kk] = v.z;
      Blds[(cb + 3) * BST + kk] = v.w;
    }
    __syncthreads();

    const float* ar0 = &Alds[((mt0 + 0) * 16 + lp) * AST + 2 * half];
    const float* ar1 = &Alds[((mt0 + 1) * 16 + lp) * AST + 2 * half];
    const float* br0 = &Blds[((nt0 + 0) * 16 + lp) * BST + 2 * half];
    const float* br1 = &Blds[((nt0 + 1) * 16 + lp) * BST + 2 * half];
#pragma unroll 4
    for (int kk = 0; kk < KC; kk += 4) {
      v2f a0 = *(const v2f*)(ar0 + kk);
      v2f a1 = *(const v2f*)(ar1 + kk);
      v2f b0 = *(const v2f*)(br0 + kk);
      v2f b1 = *(const v2f*)(br1 + kk);
      acc[0][0] = __builtin_amdgcn_wmma_f32_16x16x4_f32(false, a0, false, b0, (short)0, acc[0][0], false, false);
      acc[0][1] = __builtin_amdgcn_wmma_f32_16x16x4_f32(false, a0, false, b1, (short)0, acc[0][1], false, false);
      acc[1][0] = __builtin_amdgcn_wmma_f32_16x16x4_f32(false, a1, false, b0, (short)0, acc[1][0], false, false);
      acc[1][1] = __builtin_amdgcn_wmma_f32_16x16x4_f32(false, a1, false, b1, (short)0, acc[1][1], false, false);
    }
    __syncthreads();
  }

#pragma unroll
  for (int mi = 0; mi < 2; ++mi)
#pragma unroll
    for (int ni = 0; ni < 2; ++ni)
#pragma unroll
      for (int r = 0; r < 8; ++r) {
        int row = (mt0 + mi) * 16 + r + 8 * half;
        int col = (nt0 + ni) * 16 + lp;
        XG[(r0 + row) * N4 + n0 + col] = acc[mi][ni][r] + bias[n0 + col];
      }
}

// ---------------------------------------------------------------------------
// Per-timestep LSTM cell step for one layer.
//   gates = [optional gbase] + concat-A[64, K] @ W[K, 4096]  (+ bias if no gbase)
// Grid: 64 blocks, block b owns gate columns [16b, 16b+16) of each f/i/g/o chunk.
// 4 waves, each with a 2x2 register tile (2 M-tiles x 2 chunks).
// ---------------------------------------------------------------------------
__global__ __launch_bounds__(128) void lstm_step_kernel(
    const float* __restrict__ gbase,   // [64, 4096] precomputed x@Wx+b, or null
    const float* __restrict__ a0,      // A rows, K-range [0, 1024)
    const float* __restrict__ a1,      // A rows, K-range [1024, 2048) (null if K==1024)
    int K,
    float*       __restrict__ c_state, // [64, 1024] in/out (column-local)
    const float* __restrict__ W,       // [K, 4096]
    const float* __restrict__ bias,    // [4096], used when gbase == null
    float*       __restrict__ h_out,   // [64, 1024]
    float*       __restrict__ h_out2)  // optional second destination, may be null
{
  __shared__ float Alds[64 * AST];
  __shared__ float Blds[64 * BST];      // transposed: [gate-col][k]
  __shared__ float Glds[4 * 64 * GST];  // [chunk][row][16] (padded)

  const int tid  = threadIdx.x;
  const int lane = tid & 31;
  const int wave = tid >> 5;          // 0..3
  const int half = lane >> 4;
  const int lp   = lane & 15;
  const int mt0  = (wave & 1) * 2;    // M tiles {mt0, mt0+1}
  const int ct0  = (wave >> 1) * 2;   // chunks {ct0, ct0+1}
  const int j0   = blockIdx.x * 16;

  v8f acc[2][2] = {};

  for (int kb = 0; kb < K; kb += KC) {
    const float* src = (kb < HID) ? (a0 + kb) : (a1 + (kb - HID));
    for (int i = tid; i < 64 * (KC / 4); i += 128) {
      int row = i >> 5, c4 = i & 31;
      float4 v = *(const float4*)(src + (size_t)row * HID + c4 * 4);
      *(float4*)(&Alds[row * AST + c4 * 4]) = v;
    }
    for (int i = tid; i < KC * 16; i += 128) {
      int kk    = i >> 4;
      int c4    = i & 15;
      int chunk = c4 >> 2;
      int col   = (c4 & 3) * 4;
      float4 v = *(const float4*)(W + (size_t)(kb + kk) * N4 + chunk * HID + j0 + col);
      int cb = chunk * 16 + col;
      Blds[(cb + 0) * BST + kk] = v.x;
      Blds[(cb + 1) * BST + kk] = v.y;
      Blds[(cb + 2) * BST + kk] = v.z;
      Blds[(cb + 3) * BST + kk] = v.w;
    }
    __syncthreads();

    const float* ar0 = &Alds[((mt0 + 0) * 16 + lp) * AST + 2 * half];
    const float* ar1 = &Alds[((mt0 + 1) * 16 + lp) * AST + 2 * half];
    const float* br0 = &Blds[((ct0 + 0) * 16 + lp) * BST + 2 * half];
    const float* br1 = &Blds[((ct0 + 1) * 16 + lp) * BST + 2 * half];
#pragma unroll 4
    for (int kk = 0; kk < KC; kk += 4) {
      v2f va0 = *(const v2f*)(ar0 + kk);
      v2f va1 = *(const v2f*)(ar1 + kk);
      v2f vb0 = *(const v2f*)(br0 + kk);
      v2f vb1 = *(const v2f*)(br1 + kk);
      acc[0][0] = __builtin_amdgcn_wmma_f32_16x16x4_f32(false, va0, false, vb0, (short)0, acc[0][0], false, false);
      acc[0][1] = __builtin_amdgcn_wmma_f32_16x16x4_f32(false, va0, false, vb1, (short)0, acc[0][1], false, false);
      acc[1][0] = __builtin_amdgcn_wmma_f32_16x16x4_f32(false, va1, false, vb0, (short)0, acc[1][0], false, false);
      acc[1][1] = __builtin_amdgcn_wmma_f32_16x16x4_f32(false, va1, false, vb1, (short)0, acc[1][1], false, false);
    }
    __syncthreads();
  }

  // Scatter: accumulator VGPR r -> tile row r + 8*half, col lp.
#pragma unroll
  for (int mi = 0; mi < 2; ++mi)
#pragma unroll
    for (int ci = 0; ci < 2; ++ci)
#pragma unroll
      for (int r = 0; r < 8; ++r) {
        int m = (mt0 + mi) * 16 + r + 8 * half;
        Glds[(ct0 + ci) * 64 * GST + m * GST + lp] = acc[mi][ci][r];
      }
  __syncthreads();

  // Epilogue: (+gbase or +bias), activations, cell update. Chunk order f,i,g,o.
  for (int idx = tid; idx < 64 * 16; idx += 128) {
    int m  = idx >> 4;
    int j  = idx & 15;
    int gj = j0 + j;
    float f, i, g, o;
    if (gbase) {
      const float* gb = gbase + (size_t)m * N4 + gj;
      f = Glds[0 * 64 * GST + m * GST + j] + gb[0 * HID];
      i = Glds[1 * 64 * GST + m * GST + j] + gb[1 * HID];
      g = Glds[2 * 64 * GST + m * GST + j] + gb[2 * HID];
      o = Glds[3 * 64 * GST + m * GST + j] + gb[3 * HID];
    } else {
      f = Glds[0 * 64 * GST + m * GST + j] + bias[0 * HID + gj];
      i = Glds[1 * 64 * GST + m * GST + j] + bias[1 * HID + gj];
      g = Glds[2 * 64 * GST + m * GST + j] + bias[2 * HID + gj];
      o = Glds[3 * 64 * GST + m * GST + j] + bias[3 * HID + gj];
    }
    float ff = 1.0f / (1.0f + __expf(-f));
    float ii = 1.0f / (1.0f + __expf(-i));
    float gg = tanhf(g);
    float oo = 1.0f / (1.0f + __expf(-o));
    int ci = m * HID + gj;
    float cn = ff * c_state[ci] + ii * gg;
    float hn = oo * tanhf(cn);
    c_state[ci] = cn;
    h_out[ci]   = hn;
    if (h_out2) h_out2[ci] = hn;
  }
}

extern "C" void kernel_launch(void* const* d_in, const int* in_sizes, int n_in,
                              void* d_out, int out_size, void* d_ws, size_t ws_size,
                              hipStream_t stream) {
  const float* x = (const float*)d_in[0];  // [512, 64, 1024]
  const float* W = (const float*)d_in[1];  // [2, 2048, 4096]
  const float* b = (const float*)d_in[2];  // [2, 4096]

  const size_t BH = (size_t)BATCH * HID;
  float* out  = (float*)d_out;
  float* h_n  = out + (size_t)T_LEN * BH;
  float* c_n  = h_n + 2 * BH;              // live c state, [L][B][H]
  float* hbuf = (float*)d_ws;              // ping-pong h: [(parity*2 + layer)][B*H]
  float* XG   = hbuf + 4 * BH;             // optional precomputed x@Wx0+b0, [T*B, 4096]

  const size_t need = (4 * BH + (size_t)T_LEN * BATCH * N4) * sizeof(float);
  const bool precomp = (ws_size >= need);

  hipMemsetAsync(hbuf, 0, 4 * BH * sizeof(float), stream);
  hipMemsetAsync(c_n,  0, 2 * BH * sizeof(float), stream);

  const float* W0  = W;                          // [2048, 4096]
  const float* Wh0 = W + (size_t)HID * N4;       // rows 1024..2047 of layer 0
  const float* W1  = W + (size_t)K2 * N4;        // layer 1

  if (precomp) {
    dim3 g(N4 / 64, (T_LEN * BATCH) / 64);       // (64, 512)
    xw_gemm_kernel<<<g, 128, 0, stream>>>(x, W0, b, XG);
  }

  dim3 grid(HID / 16);  // 64 blocks
  dim3 block(128);

  for (int t = 0; t < T_LEN; ++t) {
    int p = t & 1;
    const float* h0_prev = hbuf + (p * 2 + 0) * BH;
    const float* h1_prev = hbuf + (p * 2 + 1) * BH;
    float* h0_new = hbuf + ((1 - p) * 2 + 0) * BH;
    float* h1_new = hbuf + ((1 - p) * 2 + 1) * BH;

    if (precomp) {
      // layer 0: x-part precomputed; only K=1024 recurrent GEMM on critical path
      lstm_step_kernel<<<grid, block, 0, stream>>>(
          XG + (size_t)t * BATCH * N4, h0_prev, nullptr, HID,
          c_n + 0 * BH, Wh0, b, h0_new, nullptr);
    } else {
      lstm_step_kernel<<<grid, block, 0, stream>>>(
          nullptr, x + (size_t)t * BH, h0_prev, K2,
          c_n + 0 * BH, W0, b, h0_new, nullptr);
    }

    // layer 1: input = h0_t (sequential, cannot precompute)
    lstm_step_kernel<<<grid, block, 0, stream>>>(
        nullptr, h0_new, h1_prev, K2,
        c_n + 1 * BH, W1, b + N4, h1_new, out + (size_t)t * BH);
  }

  // Final h state sits at parity 0, laid out [L, B, H].
  hipMemcpyAsync(h_n, hbuf, 2 * BH * sizeof(float), hipMemcpyDeviceToDevice, stream);
}